// ssa_decomposition_45629732553330
// MI455X (gfx1250) — compile-verified
//
#include <hip/hip_runtime.h>

typedef float v2f __attribute__((ext_vector_type(2)));
typedef float v8f __attribute__((ext_vector_type(8)));

#define LWIN    128
#define NCOMP   4
#define NSAMP   2048
#define KLEN    (NSAMP - LWIN + 1)   /* 1921 */
#define THREADS 256
#define NITER   64

/* LDS layout (float offsets) */
#define XS_OFF  0
#define XS_LEN  2064
#define CM_OFF  (XS_OFF + XS_LEN)     /* 128x128 Gram; later reused for a[k][j] */
#define CM_LEN  (LWIN * LWIN)
#define QM_OFF  (CM_OFF + CM_LEN)
#define QM_LEN  (LWIN * NCOMP)
#define ZM_OFF  (QM_OFF + QM_LEN)
#define ZM_LEN  (LWIN * NCOMP)
#define GM_OFF  (ZM_OFF + ZM_LEN)
#define GM_LEN  16
#define RM_OFF  (GM_OFF + GM_LEN)
#define RM_LEN  16
#define SMEM_FLOATS (RM_OFF + RM_LEN)

__global__ void __launch_bounds__(THREADS)
ssa_decomp_kernel(const float* __restrict__ x, float* __restrict__ out) {
  extern __shared__ float smem[];
  float* xs = smem + XS_OFF;
  float* Cm = smem + CM_OFF;
  float* Qm = smem + QM_OFF;
  float* Zm = smem + ZM_OFF;
  float* Gm = smem + GM_OFF;
  float* Rm = smem + RM_OFF;

  const int tid = threadIdx.x;
  const int bc  = blockIdx.x;   /* 0..63 == b*8 + c */

  /* ---- Phase 1: stage signal into LDS ---- */
  for (int i = tid; i < XS_LEN; i += THREADS)
    xs[i] = (i < NSAMP) ? x[(size_t)bc * NSAMP + i] : 0.0f;
  __syncthreads();

  /* ---- Phase 2: Gram matrix C = traj^T * traj via V_WMMA_F32_16X16X4_F32 ----
     C[l1,l2] = sum_k x[k+l1]*x[k+l2], k in [0,1921).
     Wave w owns row block l1 = 16*w; sweeps 8 column blocks.
     WMMA f32 16x16x4 layouts (wave32):
       A 16x4:  lane<16 -> M=lane, VGPR0/1 = K0/K1; lane>=16 -> M=lane-16, K2/K3
       B 4x16:  lane<16 -> N=lane, VGPR0/1 = K0/K1; lane>=16 -> N=lane-16, K2/K3
       C/D 16x16: VGPR i: M=i (lanes 0-15), M=i+8 (lanes 16-31), N=lane&15   */
  {
    const int wv   = tid >> 5;
    const int lane = tid & 31;
    const int hi   = lane >> 4;
    const int ln   = lane & 15;
    const int l1   = 16 * wv;

    v8f acc[8];
    #pragma unroll
    for (int j = 0; j < 8; ++j)
      #pragma unroll
      for (int i = 0; i < 8; ++i) acc[j][i] = 0.0f;

    for (int k0 = 0; k0 < 1920; k0 += 4) {
      const int base = k0 + 2 * hi + ln;
      v2f aA;
      aA.x = xs[base + l1];
      aA.y = xs[base + l1 + 1];
      #pragma unroll
      for (int j = 0; j < 8; ++j) {
        v2f bB;
        bB.x = xs[base + 16 * j];
        bB.y = xs[base + 16 * j + 1];
        acc[j] = __builtin_amdgcn_wmma_f32_16x16x4_f32(
            false, aA, false, bB, (short)0, acc[j], false, false);
      }
    }
    /* tail term k = 1920 (K = 480*4 + 1) done on VALU in C-layout */
    float av[8];
    #pragma unroll
    for (int i = 0; i < 8; ++i) av[i] = xs[1920 + l1 + 8 * hi + i];
    #pragma unroll
    for (int j = 0; j < 8; ++j) {
      const float bs = xs[1920 + 16 * j + ln];
      #pragma unroll
      for (int i = 0; i < 8; ++i) acc[j][i] += av[i] * bs;
    }
    /* scatter C tiles to LDS */
    #pragma unroll
    for (int j = 0; j < 8; ++j)
      #pragma unroll
      for (int i = 0; i < 8; ++i)
        Cm[(l1 + 8 * hi + i) * LWIN + 16 * j + ln] = acc[j][i];
  }

  /* deterministic pseudo-random init of Q (128x4) */
  if (tid < LWIN) {
    #pragma unroll
    for (int j = 0; j < NCOMP; ++j) {
      unsigned h = (unsigned)(tid * NCOMP + j) * 2654435761u + 12345u;
      h ^= h >> 16; h *= 2246822519u; h ^= h >> 13;
      Qm[tid * NCOMP + j] = (float)(h & 0xFFFFFFu) * (1.0f / 8388608.0f) - 1.0f;
    }
  }
  __syncthreads();

  /* ---- Phase 3: orthogonal iteration with Cholesky-QR -> top-4 eigvecs ---- */
  for (int it = 0; it < NITER; ++it) {
    float z[NCOMP] = {0.f, 0.f, 0.f, 0.f};
    if (tid < LWIN) {
      for (int m = 0; m < LWIN; ++m) {
        const float cv = Cm[tid * LWIN + m];
        #pragma unroll
        for (int j = 0; j < NCOMP; ++j) z[j] += cv * Qm[m * NCOMP + j];
      }
      #pragma unroll
      for (int j = 0; j < NCOMP; ++j) Zm[tid * NCOMP + j] = z[j];
    }
    __syncthreads();
    /* G = Z^T Z (upper triangle) */
    if (tid < 16) {
      const int j0 = tid >> 2, j1 = tid & 3;
      float s = 0.f;
      if (j0 <= j1)
        for (int m = 0; m < LWIN; ++m)
          s += Zm[m * NCOMP + j0] * Zm[m * NCOMP + j1];
      Gm[tid] = s;
    }
    __syncthreads();
    /* thread 0: 4x4 Cholesky G = R^T R, then Rinv (upper triangular) */
    if (tid == 0) {
      float g[4][4], R[4][4], Ri[4][4];
      for (int i2 = 0; i2 < 4; ++i2)
        for (int j2 = 0; j2 < 4; ++j2)
          g[i2][j2] = (i2 <= j2) ? Gm[i2 * 4 + j2] : Gm[j2 * 4 + i2];
      for (int i2 = 0; i2 < 4; ++i2) {
        float s = g[i2][i2];
        for (int k2 = 0; k2 < i2; ++k2) s -= R[k2][i2] * R[k2][i2];
        s = fmaxf(s, 1e-30f);
        R[i2][i2] = sqrtf(s);
        const float inv = 1.0f / R[i2][i2];
        for (int j2 = i2 + 1; j2 < 4; ++j2) {
          float t = g[i2][j2];
          for (int k2 = 0; k2 < i2; ++k2) t -= R[k2][i2] * R[k2][j2];
          R[i2][j2] = t * inv;
        }
      }
      for (int i2 = 0; i2 < 4; ++i2)
        for (int j2 = 0; j2 < 4; ++j2) Ri[i2][j2] = 0.f;
      for (int i2 = 3; i2 >= 0; --i2) {
        Ri[i2][i2] = 1.0f / R[i2][i2];
        for (int j2 = i2 + 1; j2 < 4; ++j2) {
          float s = 0.f;
          for (int k2 = i2 + 1; k2 <= j2; ++k2) s += R[i2][k2] * Ri[k2][j2];
          Ri[i2][j2] = -s * Ri[i2][i2];
        }
      }
      for (int i2 = 0; i2 < 4; ++i2)
        for (int j2 = 0; j2 < 4; ++j2) Rm[i2 * 4 + j2] = Ri[i2][j2];
    }
    __syncthreads();
    /* Q = Z * Rinv */
    if (tid < LWIN) {
      float qn[NCOMP];
      #pragma unroll
      for (int j = 0; j < NCOMP; ++j) {
        float s = 0.f;
        #pragma unroll
        for (int k2 = 0; k2 < NCOMP; ++k2)
          if (k2 <= j) s += z[k2] * Rm[k2 * 4 + j];
        qn[j] = s;
      }
      #pragma unroll
      for (int j = 0; j < NCOMP; ++j) Qm[tid * NCOMP + j] = qn[j];
    }
    __syncthreads();
  }

  /* ---- Phase 4: a[k][j] = sum_l x[k+l] * v_j[l]  (reuse Cm LDS) ---- */
  float* am = Cm;   /* KLEN*NCOMP = 7684 floats < 16384 */
  for (int k = tid; k < KLEN; k += THREADS) {
    float s[NCOMP] = {0.f, 0.f, 0.f, 0.f};
    for (int l = 0; l < LWIN; ++l) {
      const float xv = xs[k + l];
      #pragma unroll
      for (int j = 0; j < NCOMP; ++j) s[j] += xv * Qm[l * NCOMP + j];
    }
    #pragma unroll
    for (int j = 0; j < NCOMP; ++j) am[k * NCOMP + j] = s[j];
  }
  __syncthreads();

  /* ---- Phase 5: anti-diagonal (Hankel) averaging -> comps[j][bc][n] ---- */
  for (int n = tid; n < NSAMP; n += THREADS) {
    const int lmin = (n - (KLEN - 1)) > 0 ? (n - (KLEN - 1)) : 0;
    const int lmax = (n < (LWIN - 1)) ? n : (LWIN - 1);
    float s[NCOMP] = {0.f, 0.f, 0.f, 0.f};
    for (int l = lmin; l <= lmax; ++l) {
      const int k = n - l;
      #pragma unroll
      for (int j = 0; j < NCOMP; ++j)
        s[j] += Qm[l * NCOMP + j] * am[k * NCOMP + j];
    }
    const float inv = 1.0f / (float)(lmax - lmin + 1);
    #pragma unroll
    for (int j = 0; j < NCOMP; ++j)
      out[((size_t)j * 64 + bc) * NSAMP + n] = s[j] * inv;
  }
}

extern "C" void kernel_launch(void* const* d_in, const int* in_sizes, int n_in,
                              void* d_out, int out_size, void* d_ws, size_t ws_size,
                              hipStream_t stream) {
  (void)in_sizes; (void)n_in; (void)out_size; (void)d_ws; (void)ws_size;
  const float* x = (const float*)d_in[0];
  float* out = (float*)d_out;
  const size_t shmem = (size_t)SMEM_FLOATS * sizeof(float);  /* ~78 KB */
  hipLaunchKernelGGL(ssa_decomp_kernel, dim3(64), dim3(THREADS), shmem, stream,
                     x, out);
}